// DyConv_37417755082932
// MI455X (gfx1250) — compile-verified
//
#include <hip/hip_runtime.h>
#include <hip/hip_bf16.h>
#include <stdint.h>

// ---------------------------------------------------------------------------
// Per-pixel dynamic (involution) conv on gfx1250 via banded-B WMMA f32.
//   out[c,p] = sum_{i,q} x[c, h-3+i, w0-3+q] * B_i[q,p],
//   B_i[q,p] = weight[n,0,i,q-p,h,w0+p] if 0<=q-p<7 else 0   (banded 24x16)
// One wave computes a 16(channel) x 16(pixel) f32 tile with
// V_WMMA_F32_16X16X4_F32, K-chunked 6x4 cols x 7 kh rows = 42 WMMAs/tile.
// x strip + raw weights staged via global_load_async_to_lds_b32 (ASYNCcnt).
// ---------------------------------------------------------------------------

typedef float v2f __attribute__((ext_vector_type(2)));
typedef float v8f __attribute__((ext_vector_type(8)));

#define Nn   8
#define Cc   256
#define Hh   128
#define Ww   128
#define KH   7
#define KW   7
#define PADm 3

#define WPT    16   // output pixels per tile (WMMA N)
#define CBLK   64   // channels per workgroup (4 waves x 16)
#define NWAVE  4
#define XCOLS  24   // staged x columns (22 needed, padded to 6 K=4 chunks)
#define XPITCH 25   // col pitch (odd -> conflict-free A-fragment reads)
#define BPITCH 17   // p pitch   (odd -> conflict-free B-fragment reads)

__global__ __launch_bounds__(128)
void dyconv_wmma_f32(const float* __restrict__ x,
                     const float* __restrict__ wgt,
                     float* __restrict__ out)
{
    __shared__ float lds_x[KH * CBLK * XPITCH];   // [i][c][q]
    __shared__ float lds_wraw[KH * KW * WPT];     // [i][j][p]
    __shared__ float lds_b[KH * XCOLS * BPITCH];  // [i][q][p] banded, zero-filled

    // blockIdx.x = ((n*128 + h)*8 + wb)*4 + cblk
    int bid  = blockIdx.x;
    int cblk = bid & 3;  bid >>= 2;
    int wb   = bid & 7;  bid >>= 3;
    int h    = bid & 127;
    int n    = bid >> 7;
    int w0   = wb * WPT;
    int cb   = cblk * CBLK;
    int tid  = threadIdx.x;

    const float* xn = x   + (size_t)n * Cc * Hh * Ww;
    const float* wn = wgt + (size_t)n * KH * KW * Hh * Ww;

    // ---- Phase 1: async-stage x strip and raw weights into LDS ------------
    // x strip: rows h-3..h+3, cols w0-3..w0+20, channels cb..cb+63
    for (int e = tid; e < KH * CBLK * XCOLS; e += 128) {
        int i = e / (CBLK * XCOLS);
        int r = e - i * (CBLK * XCOLS);
        int c = r / XCOLS;
        int q = r - c * XCOLS;
        int row = h - PADm + i;
        int col = w0 - PADm + q;
        float* lp = &lds_x[(i * CBLK + c) * XPITCH + q];
        if ((unsigned)row < (unsigned)Hh && (unsigned)col < (unsigned)Ww) {
            const float* gp = xn + ((size_t)(cb + c) * Hh + row) * Ww + col;
            asm volatile("global_load_async_to_lds_b32 %0, %1, off"
                         :: "v"((uint32_t)(uintptr_t)lp),
                            "v"((uint64_t)(uintptr_t)gp)
                         : "memory");
        } else {
            *lp = 0.0f;  // zero-pad halo (DS store, disjoint from async targets)
        }
    }
    // raw weights for this (n,h,w0): [i][j][p], p-contiguous -> coalesced
    for (int e = tid; e < KH * KW * WPT; e += 128) {
        int i = e / (KW * WPT);
        int r = e - i * (KW * WPT);
        int j = r / WPT;
        int p = r - j * WPT;
        const float* gp = wn + (((size_t)(i * KW + j)) * Hh + h) * Ww + (w0 + p);
        asm volatile("global_load_async_to_lds_b32 %0, %1, off"
                     :: "v"((uint32_t)(uintptr_t)&lds_wraw[e]),
                        "v"((uint64_t)(uintptr_t)gp)
                     : "memory");
    }
    asm volatile("s_wait_asynccnt 0" ::: "memory");
    __syncthreads();

    // ---- Phase 2: expand banded B in LDS (kills all inner-loop predication)
    for (int e = tid; e < KH * XCOLS * WPT; e += 128) {
        int i = e / (XCOLS * WPT);
        int r = e - i * (XCOLS * WPT);
        int q = r / WPT;
        int p = r - q * WPT;
        int j = q - p;
        float v = 0.0f;
        if ((unsigned)j < (unsigned)KW) v = lds_wraw[(i * KW + j) * WPT + p];
        lds_b[(i * XCOLS + q) * BPITCH + p] = v;
    }
    __syncthreads();

    // ---- Phase 3: 42 x V_WMMA_F32_16X16X4_F32 per wave --------------------
    int lane = tid & 31;
    int wv   = tid >> 5;        // wave -> 16-channel sub-tile
    int m    = lane & 15;       // A row (channel) == B col (pixel) per layout
    int half = lane >> 4;       // K split: lanes 16-31 hold K=2,3

    v8f acc = {};
    int crow = WPT * wv + m;    // channel within CBLK for A-fragment reads

    for (int i = 0; i < KH; ++i) {
        const float* xrow = &lds_x[(i * CBLK + crow) * XPITCH];
        const float* brow = &lds_b[i * XCOLS * BPITCH];
        #pragma unroll
        for (int ch = 0; ch < XCOLS / 4; ++ch) {
            int qa = ch * 4 + half * 2;
            v2f a, b;
            a.x = xrow[qa];                   // A[m, qa]   (16x4 layout)
            a.y = xrow[qa + 1];               // A[m, qa+1]
            b.x = brow[qa * BPITCH + m];      // B[qa, p=m] (4x16 layout)
            b.y = brow[(qa + 1) * BPITCH + m];
            acc = __builtin_amdgcn_wmma_f32_16x16x4_f32(
                false, a, false, b, (short)0, acc, false, false);
        }
    }

    // ---- Store: D lane layout -> out[n, cb+16*wv + v + 8*half, h, w0+m] ---
    #pragma unroll
    for (int v = 0; v < 8; ++v) {
        int c = cb + WPT * wv + v + 8 * half;
        out[(((size_t)n * Cc + c) * Hh + h) * Ww + (w0 + m)] = acc[v];
    }
}

extern "C" void kernel_launch(void* const* d_in, const int* in_sizes, int n_in,
                              void* d_out, int out_size, void* d_ws, size_t ws_size,
                              hipStream_t stream) {
    (void)in_sizes; (void)n_in; (void)out_size; (void)d_ws; (void)ws_size;
    const float* x = (const float*)d_in[0];
    const float* w = (const float*)d_in[1];
    float* out = (float*)d_out;
    // grid = N * H * (W/16) * (C/64) = 8*128*8*4
    dim3 grid(Nn * Hh * (Ww / WPT) * (Cc / CBLK));
    dyconv_wmma_f32<<<grid, NWAVE * 32, 0, stream>>>(x, w, out);
}